// HMMNet_26319559590582
// MI455X (gfx1250) — compile-verified
//
#include <hip/hip_runtime.h>
#include <math.h>

// Problem constants (match reference)
#define TSTEPS 65536
#define BSTATE 64
#define AACT   32
// Scan chunking: 512 chunks x 128 steps covers the 65535 transition steps.
#define NCHUNK 512
#define CHUNKL 128
#define RESCALE_EVERY 8
#define LN2F 0.69314718055994530942f

typedef float v2f __attribute__((ext_vector_type(2)));
typedef float v8f __attribute__((ext_vector_type(8)));

__device__ __forceinline__ float wave_max_f32(float v) {
#pragma unroll
  for (int off = 16; off > 0; off >>= 1)
    v = fmaxf(v, __shfl_xor(v, off, 32));
  return v;
}

// Issue async global->LDS staging of the raw per-step log values for step t.
// Per lane: stop (2 floats, b64), start (1 float, b32), action gather (b32).
// GVS addressing: 64-bit SGPR base + 32-bit per-lane VGPR byte offset.
__device__ __forceinline__ void async_stage(
    const float* __restrict__ act, const float* __restrict__ stop,
    const float* __restrict__ start, const int* __restrict__ actions, int t,
    int tid, float2* rstop, float* rstart, float* ract) {
  const int a = actions[t];
  const unsigned ds_stop  = (unsigned)(size_t)(rstop + tid);
  const unsigned ds_start = (unsigned)(size_t)(rstart + tid);
  const unsigned ds_act   = (unsigned)(size_t)(ract + tid);
  const unsigned vo_stop  = (unsigned)((t * BSTATE + tid) * 2) * 4u;
  const unsigned vo_start = (unsigned)(t * BSTATE + tid) * 4u;
  const unsigned vo_act   = (unsigned)((t * BSTATE + tid) * AACT + a) * 4u;
  asm volatile("global_load_async_to_lds_b64 %0, %1, %2"
               :: "v"(ds_stop), "v"(vo_stop), "s"(stop) : "memory");
  asm volatile("global_load_async_to_lds_b32 %0, %1, %2"
               :: "v"(ds_start), "v"(vo_start), "s"(start) : "memory");
  asm volatile("global_load_async_to_lds_b32 %0, %1, %2"
               :: "v"(ds_act), "v"(vo_act), "s"(act) : "memory");
}

// ---------------------------------------------------------------------------
// Pass 1: each block composes one chunk of step operators Q_t = b*(s.e)^T + D
// into a 64x64 f32 matrix, exploiting the rank-1 + diagonal structure:
//   M <- (M*b) * (s.e)^T + M .* (d.e)   (O(B^2) per step)
// Thread i holds row i of M in 64 VGPRs. Per-step column factors live in LDS.
// Raw logp inputs for step t+1 are double-buffered into LDS with
// GLOBAL_LOAD_ASYNC_TO_LDS while step t computes (ASYNCcnt-synchronized).
// Periodic power-of-2 rescaling keeps values in f32 range; log-scale tracked.
// ---------------------------------------------------------------------------
__global__ __launch_bounds__(BSTATE) void hmm_chunk_compose(
    const float* __restrict__ act, const float* __restrict__ stop,
    const float* __restrict__ start, const int* __restrict__ actions,
    float* __restrict__ mats, float* __restrict__ scales) {
  __shared__ float  lb[BSTATE];    // b[k] = exp(stop_logp)
  __shared__ float2 lsd[BSTATE];   // (s[j]*e[j], d[j]*e[j])
  __shared__ float2 rstop[2][BSTATE];   // raw staging (double buffered)
  __shared__ float  rstart[2][BSTATE];
  __shared__ float  ract[2][BSTATE];
  __shared__ float  lred[2];

  const int tid = threadIdx.x;  // row index i
  const int c = blockIdx.x;
  const int t0 = 1 + c * CHUNKL;
  const int t1 = min(t0 + CHUNKL, TSTEPS);  // transition steps t in [t0, t1)

  float row[BSTATE];
#pragma unroll
  for (int j = 0; j < BSTATE; ++j) row[j] = (j == tid) ? 1.f : 0.f;
  float logscale = 0.f;
  int since = 0;

  // Prologue: stage step t0 into buffer 1 (t0 is odd).
  if (t0 < t1)
    async_stage(act, stop, start, actions, t0, tid, rstop[t0 & 1],
                rstart[t0 & 1], ract[t0 & 1]);

  for (int t = t0; t < t1; ++t) {
    const int buf = t & 1;
    // Stage next step into the other buffer, then drain this buffer's 3
    // transfers (async loads complete in order => waiting to <=3 is enough).
    if (t + 1 < t1) {
      async_stage(act, stop, start, actions, t + 1, tid, rstop[buf ^ 1],
                  rstart[buf ^ 1], ract[buf ^ 1]);
      asm volatile("s_wait_asynccnt 0x3" ::: "memory");
    } else {
      asm volatile("s_wait_asynccnt 0x0" ::: "memory");
    }
    __syncthreads();  // make all lanes' async LDS writes visible to all waves

    // Owner thread builds this step's column factors from the staged raws.
    const float2 sp = rstop[buf][tid];       // (beta, cont)
    const float st  = rstart[buf][tid];
    const float aj  = ract[buf][tid];
    lb[tid] = __expf(sp.x);
    lsd[tid] = make_float2(__expf(st + aj), __expf(sp.y + aj));
    __syncthreads();

    // v = row_i . b   (4 split accumulators to break the FMA chain)
    float v0 = 0.f, v1 = 0.f, v2 = 0.f, v3 = 0.f;
#pragma unroll
    for (int j = 0; j < BSTATE; j += 4) {
      v0 = fmaf(row[j + 0], lb[j + 0], v0);
      v1 = fmaf(row[j + 1], lb[j + 1], v1);
      v2 = fmaf(row[j + 2], lb[j + 2], v2);
      v3 = fmaf(row[j + 3], lb[j + 3], v3);
    }
    const float v = (v0 + v1) + (v2 + v3);
#pragma unroll
    for (int j = 0; j < BSTATE; ++j) {
      const float2 sd = lsd[j];
      row[j] = fmaf(v, sd.x, row[j] * sd.y);
    }

    if (++since == RESCALE_EVERY) {
      since = 0;
      float m0 = 0.f, m1 = 0.f, m2 = 0.f, m3 = 0.f;
#pragma unroll
      for (int j = 0; j < BSTATE; j += 4) {
        m0 = fmaxf(m0, row[j + 0]);
        m1 = fmaxf(m1, row[j + 1]);
        m2 = fmaxf(m2, row[j + 2]);
        m3 = fmaxf(m3, row[j + 3]);
      }
      float mx = wave_max_f32(fmaxf(fmaxf(m0, m1), fmaxf(m2, m3)));
      if ((tid & 31) == 0) lred[tid >> 5] = mx;
      __syncthreads();
      mx = fmaxf(lred[0], lred[1]);
      if (mx > 0.f) {
        int ex;
        (void)frexpf(mx, &ex);
        const float sc = exp2f((float)(-ex));
#pragma unroll
        for (int j = 0; j < BSTATE; ++j) row[j] *= sc;
        logscale += (float)ex * LN2F;
      }
      __syncthreads();
    }
  }

  float* o = mats + (size_t)c * (BSTATE * BSTATE) + (size_t)tid * BSTATE;
#pragma unroll
  for (int j = 0; j < BSTATE; j += 4)
    *(float4*)(o + j) = make_float4(row[j], row[j + 1], row[j + 2], row[j + 3]);
  if (tid == 0) scales[c] = logscale;
}

// ---------------------------------------------------------------------------
// Pass 2: pairwise tree combine.  D = M_{2p} x M_{2p+1}  (64x64x64, f32),
// computed with V_WMMA_F32_16X16X4_F32.  4 waves per block, each wave owns a
// 16-row band (4 output tiles, 16 k-steps each => 64 wmma per block-wave set).
// Result is rescaled by its block max; log-scales accumulate.
// ---------------------------------------------------------------------------
__global__ __launch_bounds__(128) void hmm_pair_combine(
    const float* __restrict__ src, const float* __restrict__ sscale,
    float* __restrict__ dst, float* __restrict__ dscale) {
  __shared__ float lred[4];
  const int p = blockIdx.x;
  const int wave = threadIdx.x >> 5;
  const int lane = threadIdx.x & 31;
  const int l15 = lane & 15;
  const int hh = lane >> 4;  // lane half select
  const float* Am = src + (size_t)(2 * p) * (64 * 64);
  const float* Bm = src + (size_t)(2 * p + 1) * (64 * 64);
  float* Dm = dst + (size_t)p * (64 * 64);
  const int r0 = wave * 16;

  v8f acc[4] = {};
#pragma unroll
  for (int n = 0; n < 4; ++n) {
#pragma unroll
    for (int k0 = 0; k0 < 64; k0 += 4) {
      // A fragment (16x4): lane l15 = row, VGPR0/1 hold K = {0,1} (+2 hi half)
      v2f a, b;
      a.x = Am[(r0 + l15) * 64 + k0 + 2 * hh];
      a.y = Am[(r0 + l15) * 64 + k0 + 2 * hh + 1];
      // B fragment (4x16): lane l15 = col, VGPR0/1 hold K = {0,1} (+2 hi half)
      b.x = Bm[(k0 + 2 * hh) * 64 + n * 16 + l15];
      b.y = Bm[(k0 + 2 * hh + 1) * 64 + n * 16 + l15];
      acc[n] = __builtin_amdgcn_wmma_f32_16x16x4_f32(
          /*neg_a=*/false, a, /*neg_b=*/false, b,
          /*c_mod=*/(short)0, acc[n], /*reuse_a=*/false, /*reuse_b=*/false);
    }
  }

  // Block-wide max for power-of-2 rescale.
  float mx = 0.f;
#pragma unroll
  for (int n = 0; n < 4; ++n)
#pragma unroll
    for (int r = 0; r < 8; ++r) mx = fmaxf(mx, acc[n][r]);
  mx = wave_max_f32(mx);
  if (lane == 0) lred[wave] = mx;
  __syncthreads();
  mx = fmaxf(fmaxf(lred[0], lred[1]), fmaxf(lred[2], lred[3]));
  int ex = 0;
  if (mx > 0.f) (void)frexpf(mx, &ex);
  const float sc = exp2f((float)(-ex));

  // C/D layout: VGPR r holds rows r (lanes 0-15) and r+8 (lanes 16-31).
#pragma unroll
  for (int n = 0; n < 4; ++n)
#pragma unroll
    for (int r = 0; r < 8; ++r)
      Dm[(r0 + r + 8 * hh) * 64 + n * 16 + l15] = acc[n][r] * sc;

  if (threadIdx.x == 0)
    dscale[p] = sscale[2 * p] + sscale[2 * p + 1] + (float)ex * LN2F;
}

// ---------------------------------------------------------------------------
// Pass 3: f0^T * M_total, dot with exp(stop_T), logsumexp -> -logp.
// ---------------------------------------------------------------------------
__global__ __launch_bounds__(BSTATE) void hmm_finalize(
    const float* __restrict__ act, const float* __restrict__ stop,
    const float* __restrict__ start, const int* __restrict__ actions,
    const float* __restrict__ Mtot, const float* __restrict__ scale,
    float* __restrict__ out) {
  __shared__ float lf[BSTATE], lred[2];
  const int j = threadIdx.x;
  const int a0 = actions[0];
  const float f0 = start[j] + act[(size_t)j * AACT + a0];
  float m0 = wave_max_f32(f0);
  if ((j & 31) == 0) lred[j >> 5] = m0;
  __syncthreads();
  m0 = fmaxf(lred[0], lred[1]);
  lf[j] = __expf(f0 - m0);
  __syncthreads();

  float fj = 0.f;
#pragma unroll
  for (int i = 0; i < BSTATE; ++i) fj = fmaf(lf[i], Mtot[i * BSTATE + j], fj);
  float gj = fj * __expf(stop[((size_t)TSTEPS * BSTATE + j) * 2 + 0]);
#pragma unroll
  for (int off = 16; off > 0; off >>= 1) gj += __shfl_xor(gj, off, 32);
  __syncthreads();
  if ((j & 31) == 0) lred[j >> 5] = gj;
  __syncthreads();
  if (j == 0) {
    const float s = lred[0] + lred[1];
    out[0] = -(logf(s) + m0 + scale[0]);
  }
}

// ---------------------------------------------------------------------------
extern "C" void kernel_launch(void* const* d_in, const int* in_sizes, int n_in,
                              void* d_out, int out_size, void* d_ws,
                              size_t ws_size, hipStream_t stream) {
  const float* act     = (const float*)d_in[0];  // (T+1, B, A)
  const float* stop    = (const float*)d_in[1];  // (T+1, B, 2)
  const float* start   = (const float*)d_in[2];  // (T+1, B)
  const int*   actions = (const int*)d_in[3];    // (T,)
  float* out = (float*)d_out;

  // Workspace layout: two ping-pong matrix regions + two scale arrays.
  float* mats0 = (float*)d_ws;
  float* mats1 = mats0 + (size_t)NCHUNK * 64 * 64;
  float* sc0   = mats1 + (size_t)NCHUNK * 64 * 64;
  float* sc1   = sc0 + NCHUNK;

  hmm_chunk_compose<<<NCHUNK, BSTATE, 0, stream>>>(act, stop, start, actions,
                                                   mats0, sc0);

  const float* s = mats0;
  const float* ss = sc0;
  float* d = mats1;
  float* ds = sc1;
  int n = NCHUNK;
  while (n > 1) {
    const int np = n >> 1;
    hmm_pair_combine<<<np, 128, 0, stream>>>(s, ss, d, ds);
    const float* ts = d;
    const float* tss = ds;
    d = (float*)s;
    ds = (float*)ss;
    s = ts;
    ss = tss;
    n = np;
  }

  hmm_finalize<<<1, BSTATE, 0, stream>>>(act, stop, start, actions, s, ss, out);
}